// GCNModel_627065225855
// MI455X (gfx1250) — compile-verified
//
#include <hip/hip_runtime.h>
#include <hip/hip_bf16.h>
#include <math.h>

typedef float v2f __attribute__((ext_vector_type(2)));
typedef float v8f __attribute__((ext_vector_type(8)));

#define NN 100000      // nodes
#define NE 1600000     // edges
#define DIM 128        // in/hidden dim
#define ODIM 10        // output classes
#define NG 512         // graphs

// ---------------------------------------------------------------- init
__global__ __launch_bounds__(256) void k_init(float* __restrict__ deg,
                                              float* __restrict__ psum,
                                              float* __restrict__ pmax,
                                              float* __restrict__ counts) {
    int i = blockIdx.x * blockDim.x + threadIdx.x;
    if (i < NN) deg[i] = 0.0f;
    if (i < NG * DIM) { psum[i] = 0.0f; pmax[i] = -INFINITY; }
    if (i < NG) counts[i] = 0.0f;
}

// ---------------------------------------------------------------- degree
__global__ __launch_bounds__(256) void k_degree(const int* __restrict__ dstE,
                                                float* __restrict__ deg) {
    int e = blockIdx.x * blockDim.x + threadIdx.x;
    if (e < NE) atomicAdd(deg + dstE[e], 1.0f);
}

__global__ __launch_bounds__(256) void k_dis(float* __restrict__ deg) {
    int i = blockIdx.x * blockDim.x + threadIdx.x;
    if (i < NN) deg[i] = rsqrtf(deg[i] + 1.0f);   // +1 = self loop; deg>=1 always
}

// ---------------------------------------------------------------- fp32 WMMA GEMM
// out[N,128] = src[N,128] @ W[128,128]; each wave computes TWO 16-row tiles so
// every B fragment feeds two v_wmma ops (halves W load traffic).
__global__ __launch_bounds__(256) void k_gemm_f32(const float* __restrict__ src,
                                                  const float* __restrict__ W,
                                                  float* __restrict__ out,
                                                  int nPairs) {
    int wave = (blockIdx.x * blockDim.x + threadIdx.x) >> 5;
    if (wave >= nPairs) return;                   // whole waves exit -> EXEC all-1s for WMMA
    int lane = threadIdx.x & 31;
    int half = lane >> 4;                         // 0: lanes 0-15, 1: lanes 16-31
    int l15  = lane & 15;
    int row0 = wave << 5;                         // 32 rows per wave

    const float* arow0 = src + (size_t)(row0 + l15) * DIM + 2 * half;       // tile 0
    const float* arow1 = arow0 + (size_t)16 * DIM;                          // tile 1

    v8f acc0[8], acc1[8];
#pragma unroll
    for (int nt = 0; nt < 8; ++nt) {
        acc0[nt] = (v8f){0.f,0.f,0.f,0.f,0.f,0.f,0.f,0.f};
        acc1[nt] = (v8f){0.f,0.f,0.f,0.f,0.f,0.f,0.f,0.f};
    }

    for (int k = 0; k < DIM; k += 4) {
        v2f a0, a1;
        a0.x = arow0[k];  a0.y = arow0[k + 1];
        a1.x = arow1[k];  a1.y = arow1[k + 1];
        const float* bcol = W + (size_t)(k + 2 * half) * DIM + l15;
#pragma unroll
        for (int nt = 0; nt < 8; ++nt) {
            v2f b;
            b.x = bcol[nt * 16];
            b.y = bcol[nt * 16 + DIM];
            acc0[nt] = __builtin_amdgcn_wmma_f32_16x16x4_f32(
                false, a0, false, b, (short)0, acc0[nt], false, false);
            acc1[nt] = __builtin_amdgcn_wmma_f32_16x16x4_f32(
                false, a1, false, b, (short)0, acc1[nt], false, false);
        }
    }

    float* orow0 = out + (size_t)(row0 + 8 * half) * DIM + l15;
    float* orow1 = orow0 + (size_t)16 * DIM;
#pragma unroll
    for (int nt = 0; nt < 8; ++nt)
#pragma unroll
        for (int r = 0; r < 8; ++r) {
            orow0[(size_t)r * DIM + nt * 16] = acc0[nt][r];
            orow1[(size_t)r * DIM + nt * 16] = acc1[nt][r];
        }
}

// ---------------------------------------------------------------- zero N*128 buffer (float4)
__global__ __launch_bounds__(256) void k_zero4(float4* __restrict__ p, int n4) {
    int i = blockIdx.x * blockDim.x + threadIdx.x;
    if (i < n4) p[i] = make_float4(0.f, 0.f, 0.f, 0.f);
}

// ---------------------------------------------------------------- edge scatter: U[d] += T[s] * dis[s]*dis[d]
// one thread per (edge, float4-chunk): indices/weights loaded once per 4 atomics,
// T row gathered as b128.
__global__ __launch_bounds__(256) void k_scatter(const int* __restrict__ srcE,
                                                 const int* __restrict__ dstE,
                                                 const float* __restrict__ dis,
                                                 const float* __restrict__ T,
                                                 float* __restrict__ U) {
    long tid = (long)blockIdx.x * blockDim.x + threadIdx.x;
    long e = tid >> 5;
    int  q = (int)(tid & 31);                 // which float4 of the 128-dim row
    if (e >= NE) return;
    int s = srcE[e], d = dstE[e];
    float w = dis[s] * dis[d];
    float4 v = ((const float4*)(T + (size_t)s * DIM))[q];
    float* u = U + (size_t)d * DIM + q * 4;
    atomicAdd(u + 0, v.x * w);
    atomicAdd(u + 1, v.y * w);
    atomicAdd(u + 2, v.z * w);
    atomicAdd(u + 3, v.w * w);
}

// self loop (separate kernel launch -> plain RMW is race-free)
__global__ __launch_bounds__(256) void k_selfloop(const float* __restrict__ dis,
                                                  const float* __restrict__ T,
                                                  float* __restrict__ U) {
    long tid = (long)blockIdx.x * blockDim.x + threadIdx.x;
    long i = tid >> 5;
    int  q = (int)(tid & 31);
    if (i >= NN) return;
    float w = dis[i] * dis[i];
    float4 v = ((const float4*)(T + (size_t)i * DIM))[q];
    float4* u = (float4*)(U + (size_t)i * DIM) + q;
    float4 o = *u;
    o.x += v.x * w; o.y += v.y * w; o.z += v.z * w; o.w += v.w * w;
    *u = o;
}

// ---------------------------------------------------------------- bias (+ optional relu), float4
__global__ __launch_bounds__(256) void k_bias_act(float4* __restrict__ U,
                                                  const float4* __restrict__ b,
                                                  int doRelu) {
    long tid = (long)blockIdx.x * blockDim.x + threadIdx.x;
    if (tid >= (long)NN * (DIM / 4)) return;
    int q = (int)(tid & 31);
    float4 v = U[tid];
    float4 bb = b[q];
    v.x += bb.x; v.y += bb.y; v.z += bb.z; v.w += bb.w;
    if (doRelu) {
        v.x = fmaxf(v.x, 0.f); v.y = fmaxf(v.y, 0.f);
        v.z = fmaxf(v.z, 0.f); v.w = fmaxf(v.w, 0.f);
    }
    U[tid] = v;
}

// ---------------------------------------------------------------- pooling
__device__ __forceinline__ void atomicMaxF(float* addr, float val) {
    if (val >= 0.0f) atomicMax((int*)addr, __float_as_int(val));
    else             atomicMin((unsigned int*)addr, __float_as_uint(val));
}

__global__ __launch_bounds__(256) void k_counts(const int* __restrict__ batch,
                                                float* __restrict__ counts) {
    int i = blockIdx.x * blockDim.x + threadIdx.x;
    if (i < NN) atomicAdd(counts + batch[i], 1.0f);
}

__global__ __launch_bounds__(256) void k_pool(const int* __restrict__ batch,
                                              const float* __restrict__ U,
                                              float* __restrict__ psum,
                                              float* __restrict__ pmax) {
    long tid = (long)blockIdx.x * blockDim.x + threadIdx.x;
    long i = tid >> 5;
    int  q = (int)(tid & 31);
    if (i >= NN) return;
    int g = batch[i];
    float4 v = ((const float4*)(U + (size_t)i * DIM))[q];
    float* ps = psum + (size_t)g * DIM + q * 4;
    float* pm = pmax + (size_t)g * DIM + q * 4;
    atomicAdd(ps + 0, v.x); atomicAdd(ps + 1, v.y);
    atomicAdd(ps + 2, v.z); atomicAdd(ps + 3, v.w);
    atomicMaxF(pm + 0, v.x); atomicMaxF(pm + 1, v.y);
    atomicMaxF(pm + 2, v.z); atomicMaxF(pm + 3, v.w);
}

// ---------------------------------------------------------------- MLP head
__global__ __launch_bounds__(256) void k_head1(const float* __restrict__ psum,
                                               const float* __restrict__ pmax,
                                               const float* __restrict__ counts,
                                               const float* __restrict__ Wc1,
                                               const float* __restrict__ bc1,
                                               float* __restrict__ gh) {
    int tid = blockIdx.x * blockDim.x + threadIdx.x;   // NG*128 threads
    if (tid >= NG * DIM) return;
    int gi = tid >> 7;
    int f  = tid & 127;
    float inv = 1.0f / fmaxf(counts[gi], 1.0f);
    float acc = bc1[f];
#pragma unroll 8
    for (int j = 0; j < DIM; ++j)
        acc += (psum[(size_t)gi * DIM + j] * inv) * Wc1[(size_t)j * DIM + f];
#pragma unroll 8
    for (int j = 0; j < DIM; ++j)
        acc += pmax[(size_t)gi * DIM + j] * Wc1[(size_t)(DIM + j) * DIM + f];
    gh[tid] = fmaxf(acc, 0.0f);
}

__global__ __launch_bounds__(256) void k_head2(const float* __restrict__ gh,
                                               const float* __restrict__ Wc2,
                                               const float* __restrict__ bc2,
                                               float* __restrict__ out) {
    int tid = blockIdx.x * blockDim.x + threadIdx.x;   // NG*ODIM threads
    if (tid >= NG * ODIM) return;
    int gi = tid / ODIM;
    int o  = tid % ODIM;
    float acc = bc2[o];
#pragma unroll 8
    for (int j = 0; j < DIM; ++j)
        acc += gh[(size_t)gi * DIM + j] * Wc2[(size_t)j * ODIM + o];
    out[tid] = acc;
}

// ================================================================ launch
extern "C" void kernel_launch(void* const* d_in, const int* in_sizes, int n_in,
                              void* d_out, int out_size, void* d_ws, size_t ws_size,
                              hipStream_t stream) {
    // inputs (setup_inputs order)
    const float* x    = (const float*)d_in[0];
    const int*   eidx = (const int*)d_in[1];      // [2, NE] flat
    const int*   bat  = (const int*)d_in[2];
    const float* W1  = (const float*)d_in[4];  const float* b1  = (const float*)d_in[5];
    const float* W2  = (const float*)d_in[6];  const float* b2  = (const float*)d_in[7];
    const float* W3  = (const float*)d_in[8];  const float* b3  = (const float*)d_in[9];
    const float* Wc1 = (const float*)d_in[10]; const float* bc1 = (const float*)d_in[11];
    const float* Wc2 = (const float*)d_in[12]; const float* bc2 = (const float*)d_in[13];
    const int* srcE = eidx;
    const int* dstE = eidx + NE;

    // workspace carve-up (floats)
    float* ws     = (float*)d_ws;
    float* dis    = ws;                                   // NN
    float* T      = dis    + 100352;                      // NN*DIM (16B aligned start)
    float* U      = T      + (size_t)NN * DIM;            // NN*DIM
    float* psum   = U      + (size_t)NN * DIM;            // NG*DIM
    float* pmax   = psum   + (size_t)NG * DIM;            // NG*DIM
    float* counts = pmax   + (size_t)NG * DIM;            // NG
    float* gh     = counts + 1024;                        // NG*DIM

    const int B = 256;
    const int nPairs   = NN / 32;                         // 3125 waves x 32 rows, exact
    const int gGemm    = (nPairs * 32 + B - 1) / B;       // 391 blocks of 8 waves
    const int gNodes   = (NN + B - 1) / B;
    const int gEdges   = (NE + B - 1) / B;
    const int gNodeF4  = (int)(((long)NN * 32 + B - 1) / B);   // node x float4-chunk
    const int gEdgeF4  = (int)(((long)NE * 32 + B - 1) / B);   // edge x float4-chunk
    const int gZero4   = (NN * DIM / 4 + B - 1) / B;

    k_init  <<<gNodes, B, 0, stream>>>(dis, psum, pmax, counts);
    k_degree<<<gEdges, B, 0, stream>>>(dstE, dis);
    k_dis   <<<gNodes, B, 0, stream>>>(dis);

    const float* lin[3]  = { x,  U,  U  };
    const float* lW[3]   = { W1, W2, W3 };
    const float* lb[3]   = { b1, b2, b3 };
    for (int l = 0; l < 3; ++l) {
        k_gemm_f32<<<gGemm, B, 0, stream>>>(lin[l], lW[l], T, nPairs);
        k_zero4   <<<gZero4, B, 0, stream>>>((float4*)U, NN * DIM / 4);
        k_scatter <<<gEdgeF4, B, 0, stream>>>(srcE, dstE, dis, T, U);
        k_selfloop<<<gNodeF4, B, 0, stream>>>(dis, T, U);
        k_bias_act<<<gNodeF4, B, 0, stream>>>((float4*)U, (const float4*)lb[l], l < 2 ? 1 : 0);
    }

    k_counts<<<gNodes, B, 0, stream>>>(bat, counts);
    k_pool  <<<gNodeF4, B, 0, stream>>>(bat, U, psum, pmax);
    k_head1 <<<(NG * DIM + B - 1) / B, B, 0, stream>>>(psum, pmax, counts, Wc1, bc1, gh);
    k_head2 <<<(NG * ODIM + B - 1) / B, B, 0, stream>>>(gh, Wc2, bc2, (float*)d_out);
}